// SeqAttention_59485297049844
// MI455X (gfx1250) — compile-verified
//
#include <hip/hip_runtime.h>

// ---------------------------------------------------------------------------
// Transformer-XL relative attention layer for MI455X (gfx1250), f32 WMMA path.
// All matmuls use V_WMMA_F32_16X16X4_F32 (full f32 precision, f32 accumulate).
//
// ws layout (floats):  w_heads[4096*3072] | r_k[1024*1024] | vec[4096*1024]
//                      | attn_out[4096*1024]   => ~84 MB total
// ---------------------------------------------------------------------------

typedef float v2f __attribute__((ext_vector_type(2)));
typedef float v8f __attribute__((ext_vector_type(8)));

#define QLEN 1024
#define BSZ 4
#define DMODEL 1024
#define NHEAD 16
#define DHEAD 64
#define WHLD (3 * NHEAD * DHEAD) /* 3072 */
#define TOK (QLEN * BSZ)         /* 4096 */

__device__ __forceinline__ v8f wmma4(v2f a, v2f b, v8f c) {
  // D = A(16x4) * B(4x16) + C(16x16), all f32
  return __builtin_amdgcn_wmma_f32_16x16x4_f32(false, a, false, b, (short)0, c,
                                               false, false);
}

// ---------------------------------------------------------------------------
// Generic row-major f32 GEMM: C[M,N] = A[M,K] @ B[K,N].
// grid = (N/256, M/64), block = 128 (4 waves).
// Each wave computes a 64x64 register-blocked tile (4 M-frags x 4 N-frags =
// 16 WMMA per k-step; 4 b64 + 8 b32 loads per 16 WMMA).
// __launch_bounds__(128, 1): ~160 live VGPRs (128 accum) -- needs the full
// 256-VGPR budget of a single wave/SIMD to stay spill-free.
// A-frag m: lane l holds A[i0 + m*16 + (l&15)][k + 2*(l>>4) .. +1]
// B-frag t: lane l holds B[k + 2*(l>>4) .. +1][j0 + t*16 + (l&15)]
// ---------------------------------------------------------------------------
__global__ __launch_bounds__(128, 1)
void gemm_f32(const float* __restrict__ A, const float* __restrict__ B,
              float* __restrict__ C, int K, int lda, int ldb, int ldc) {
  const int wave = threadIdx.x >> 5;
  const int lane = threadIdx.x & 31;
  const int r = lane & 15;
  const int h = lane >> 4;
  const int i0 = blockIdx.y * 64;
  const int j0 = blockIdx.x * 256 + wave * 64;

  v8f acc[4][4];
#pragma unroll
  for (int m = 0; m < 4; ++m)
#pragma unroll
    for (int t = 0; t < 4; ++t) acc[m][t] = (v8f){};

  const float* Ap = A + (size_t)(i0 + r) * lda + 2 * h;

  for (int k = 0; k < K; k += 4) {
    v2f a[4];
#pragma unroll
    for (int m = 0; m < 4; ++m)
      a[m] = *(const v2f*)(Ap + (size_t)m * 16 * lda + k);

    const float* Bp = B + (size_t)(k + 2 * h) * ldb + j0 + r;
#pragma unroll
    for (int t = 0; t < 4; ++t) {
      v2f bf;
      bf.x = Bp[t * 16];
      bf.y = Bp[ldb + t * 16];
#pragma unroll
      for (int m = 0; m < 4; ++m) acc[m][t] = wmma4(a[m], bf, acc[m][t]);
    }
  }

#pragma unroll
  for (int m = 0; m < 4; ++m) {
#pragma unroll
    for (int v = 0; v < 8; ++v) {
      int row = i0 + m * 16 + v + h * 8;  // C layout: lanes 0-15 M=v, 16-31 M=v+8
      float* Cp = C + (size_t)row * ldc + j0 + r;
#pragma unroll
      for (int t = 0; t < 4; ++t) Cp[t * 16] = acc[m][t][v];
    }
  }
}

// ---------------------------------------------------------------------------
// Fused rel-attention: scores = (Qw@K^T + rel_shift(Qr@Rk^T)) * 1/8, causal
// mask, online softmax, O = P@V.  One wave handles a 16-row query tile for one
// (batch, head).  grid = (QLEN/16, NHEAD, BSZ), block = 32.
//
// BD gather: BD[i][j] = BD_raw[i][L-1-i+j]; for tile (i0,j0) the needed
// columns are the 31-wide band starting at cA = L-16-i0+j0 (always in-range
// for causal tiles j0 <= i0).  We compute two 16x16 WMMA tiles over the band,
// spill to LDS, and gather with index 15 - row + col.
// ---------------------------------------------------------------------------
__global__ __launch_bounds__(32, 1)
void attn_kernel(const float* __restrict__ wh,  // [4096, 3072]
                 const float* __restrict__ rk,  // [1024, 1024]
                 const float* __restrict__ rwb, // r_w_bias [16*64]
                 const float* __restrict__ rrb, // r_r_bias [16*64]
                 float* __restrict__ vec) {     // [4096, 1024]
  __shared__ float lds_u[16 * 32];  // BD_raw band
  __shared__ float lds_p[16 * 16];  // score / prob tile
  __shared__ float lds_f[16];       // per-row rescale factors
  __shared__ float lds_rs[16];      // per-row softmax denominators

  const int it = blockIdx.x;
  const int n = blockIdx.y;
  const int b = blockIdx.z;
  const int lane = threadIdx.x & 31;
  const int r = lane & 15;
  const int h = lane >> 4;
  const int i0 = it * 16;
  const float scale = 0.125f;  // 1/sqrt(64)

  // Q fragments (A-layout), with r_w_bias / r_r_bias added
  v2f qw[16], qr[16];
  {
    const float* qbase = wh + ((size_t)(i0 + r) * BSZ + b) * WHLD + n * DHEAD;
    const float* bw = rwb + n * DHEAD;
    const float* br = rrb + n * DHEAD;
#pragma unroll
    for (int kk = 0; kk < 16; ++kk) {
      int kd = kk * 4 + 2 * h;
      float q0 = qbase[kd], q1 = qbase[kd + 1];
      qw[kk].x = q0 + bw[kd]; qw[kk].y = q1 + bw[kd + 1];
      qr[kk].x = q0 + br[kd]; qr[kk].y = q1 + br[kd + 1];
    }
  }

  v8f o0 = {}, o1 = {}, o2 = {}, o3 = {};
  float m_row = -1e30f, s_row = 0.0f;  // valid in lanes 0..15 (row = lane)

  for (int jt = 0; jt <= it; ++jt) {
    const int j0 = jt * 16;
    const int cA = QLEN - 16 - i0 + j0;

    // AC tile + 2 BD_raw band tiles
    v8f acS = {}, u0 = {}, u1 = {};
    const float* kbase =
        wh + ((size_t)(j0 + r) * BSZ + b) * WHLD + NHEAD * DHEAD + n * DHEAD + 2 * h;
    const float* r0base = rk + (size_t)(cA + r) * (NHEAD * DHEAD) + n * DHEAD + 2 * h;
    const float* r1base = rk + (size_t)(cA + 16 + r) * (NHEAD * DHEAD) + n * DHEAD + 2 * h;
#pragma unroll
    for (int kk = 0; kk < 16; ++kk) {
      v2f bk = *(const v2f*)(kbase + kk * 4);
      v2f bb0 = *(const v2f*)(r0base + kk * 4);
      v2f bb1 = *(const v2f*)(r1base + kk * 4);
      acS = wmma4(qw[kk], bk, acS);
      u0 = wmma4(qr[kk], bb0, u0);
      u1 = wmma4(qr[kk], bb1, u1);
    }

    // spill BD band to LDS for the rel-shift gather
#pragma unroll
    for (int v = 0; v < 8; ++v) {
      lds_u[(v + h * 8) * 32 + r] = u0[v];
      lds_u[(v + h * 8) * 32 + 16 + r] = u1[v];
    }
    __syncthreads();

    // score tile (C layout: row = v + 8h, col = r) -> lds_p
#pragma unroll
    for (int v = 0; v < 8; ++v) {
      int row = v + h * 8;
      float bd = lds_u[row * 32 + (15 - row + r)];
      float s = (acS[v] + bd) * scale;
      if (j0 + r > i0 + row) s = -1e30f;  // causal mask
      lds_p[row * 16 + r] = s;
    }
    __syncthreads();

    // online softmax update: lanes 0..15 own one row each
    if (lane < 16) {
      int row = lane;
      float tm = m_row;
#pragma unroll
      for (int c = 0; c < 16; ++c) tm = fmaxf(tm, lds_p[row * 16 + c]);
      float f = __expf(m_row - tm);
      float ts = s_row * f;
#pragma unroll
      for (int c = 0; c < 16; ++c) {
        float p = __expf(lds_p[row * 16 + c] - tm);
        lds_p[row * 16 + c] = p;
        ts += p;
      }
      m_row = tm;
      s_row = ts;
      lds_f[row] = f;
    }
    __syncthreads();

    // rescale O accumulators (row-dependent factor)
#pragma unroll
    for (int v = 0; v < 8; ++v) {
      float f = lds_f[v + h * 8];
      o0[v] *= f; o1[v] *= f; o2[v] *= f; o3[v] *= f;
    }

    // O += P @ V   (A-frag of P read from LDS in 16x4 layout)
#pragma unroll
    for (int kk = 0; kk < 4; ++kk) {
      int kloc = kk * 4 + 2 * h;
      v2f a;
      a.x = lds_p[r * 16 + kloc];
      a.y = lds_p[r * 16 + kloc + 1];
      const float* v0p = wh + ((size_t)(j0 + kloc) * BSZ + b) * WHLD +
                         2 * NHEAD * DHEAD + n * DHEAD + r;
      const float* v1p = v0p + (size_t)BSZ * WHLD;
      v2f bv0, bv1, bv2, bv3;
      bv0.x = v0p[0];  bv0.y = v1p[0];
      bv1.x = v0p[16]; bv1.y = v1p[16];
      bv2.x = v0p[32]; bv2.y = v1p[32];
      bv3.x = v0p[48]; bv3.y = v1p[48];
      o0 = wmma4(a, bv0, o0);
      o1 = wmma4(a, bv1, o1);
      o2 = wmma4(a, bv2, o2);
      o3 = wmma4(a, bv3, o3);
    }
    __syncthreads();
  }

  // normalize and store vec[(i*BSZ+b)*1024 + n*64 + d]
  if (lane < 16) lds_rs[lane] = s_row;
  __syncthreads();
#pragma unroll
  for (int v = 0; v < 8; ++v) {
    int row = v + h * 8;
    float inv = 1.0f / lds_rs[row];
    float* op = vec + ((size_t)(i0 + row) * BSZ + b) * (NHEAD * DHEAD) + n * DHEAD + r;
    op[0]  = o0[v] * inv;
    op[16] = o1[v] * inv;
    op[32] = o2[v] * inv;
    op[48] = o3[v] * inv;
  }
}

// ---------------------------------------------------------------------------
// out = LayerNorm(w + attn_out); one 256-thread block per token row (1024 el.)
// ---------------------------------------------------------------------------
__global__ void ln_kernel(const float* __restrict__ w, const float* __restrict__ att,
                          const float* __restrict__ g, const float* __restrict__ bta,
                          float* __restrict__ out) {
  __shared__ float red[256];
  const int t = blockIdx.x;
  const int tid = threadIdx.x;
  const float* xw = w + (size_t)t * DMODEL;
  const float* xa = att + (size_t)t * DMODEL;

  float x[4];
  float s = 0.f;
#pragma unroll
  for (int u = 0; u < 4; ++u) {
    x[u] = xw[tid + u * 256] + xa[tid + u * 256];
    s += x[u];
  }
  red[tid] = s;
  __syncthreads();
  for (int off = 128; off > 0; off >>= 1) {
    if (tid < off) red[tid] += red[tid + off];
    __syncthreads();
  }
  const float mu = red[0] * (1.0f / DMODEL);
  __syncthreads();

  float vs = 0.f;
#pragma unroll
  for (int u = 0; u < 4; ++u) {
    float d = x[u] - mu;
    vs += d * d;
  }
  red[tid] = vs;
  __syncthreads();
  for (int off = 128; off > 0; off >>= 1) {
    if (tid < off) red[tid] += red[tid + off];
    __syncthreads();
  }
  const float var = red[0] * (1.0f / DMODEL);
  const float rstd = rsqrtf(var + 1e-5f);

  float* op = out + (size_t)t * DMODEL;
#pragma unroll
  for (int u = 0; u < 4; ++u) {
    int c = tid + u * 256;
    op[c] = (x[u] - mu) * rstd * g[c] + bta[c];
  }
}

// ---------------------------------------------------------------------------
extern "C" void kernel_launch(void* const* d_in, const int* in_sizes, int n_in,
                              void* d_out, int out_size, void* d_ws, size_t ws_size,
                              hipStream_t stream) {
  const float* w    = (const float*)d_in[0];  // [1024,4,1024]
  const float* rmat = (const float*)d_in[1];  // [1024,1024]
  // d_in[2] = attn_mask (causal; computed implicitly)
  const float* qkvw = (const float*)d_in[3];  // [1024,3072]
  const float* rw   = (const float*)d_in[4];  // [1024,1024]
  const float* ow   = (const float*)d_in[5];  // [1024,1024]
  const float* rrb  = (const float*)d_in[6];  // r_r_bias [16,64]
  const float* rwb  = (const float*)d_in[7];  // r_w_bias [16,64]
  const float* lng  = (const float*)d_in[8];
  const float* lnb  = (const float*)d_in[9];
  float* out = (float*)d_out;

  float* ws = (float*)d_ws;
  float* wh   = ws;                                // 4096*3072
  float* rk   = wh + (size_t)TOK * WHLD;           // 1024*1024
  float* vec  = rk + (size_t)QLEN * DMODEL;        // 4096*1024
  float* aout = vec + (size_t)TOK * DMODEL;        // 4096*1024

  // w_heads = w @ qkv_w  : [4096,1024]x[1024,3072]
  gemm_f32<<<dim3(WHLD / 256, TOK / 64), 128, 0, stream>>>(w, qkvw, wh, DMODEL,
                                                           DMODEL, WHLD, WHLD);
  // r_k = r @ r_w : [1024,1024]x[1024,1024]
  gemm_f32<<<dim3(DMODEL / 256, QLEN / 64), 128, 0, stream>>>(
      rmat, rw, rk, DMODEL, DMODEL, DMODEL, DMODEL);
  // fused relative attention
  attn_kernel<<<dim3(QLEN / 16, NHEAD, BSZ), 32, 0, stream>>>(wh, rk, rwb, rrb, vec);
  // attn_out = vec @ o_w : [4096,1024]x[1024,1024]
  gemm_f32<<<dim3(DMODEL / 256, TOK / 64), 128, 0, stream>>>(
      vec, ow, aout, DMODEL, DMODEL, DMODEL, DMODEL);
  // out = LN(w + attn_out)
  ln_kernel<<<TOK, 256, 0, stream>>>(w, aout, lng, lnb, out);
}